// AddWeightedSwappedInEdges_36816459662136
// MI455X (gfx1250) — compile-verified
//
#include <hip/hip_runtime.h>

typedef __attribute__((ext_vector_type(2))) float v2f;
typedef __attribute__((ext_vector_type(4))) float v4f;
typedef __attribute__((ext_vector_type(8))) float v8f;
typedef __attribute__((ext_vector_type(4))) int   v4i;

// Native CDNA5 fp32 global atomic add (no return -> STOREcnt path, no CAS loop).
__device__ __forceinline__ void gadd_f32(float* p, float v) {
    asm volatile("global_atomic_add_f32 %0, %1, off" : : "v"(p), "v"(v) : "memory");
}

// ---------------------------------------------------------------------------
// Pass 1: per-edge gather + weighted scatter-add into pooled[N][2].
// 4 edges per thread; the three 384MB single-use edge streams are loaded
// non-temporally (th:TH_LOAD_NT) so they don't evict the L2-resident 8MB
// hidden/pooled tables that take ~64M random hits each.
// ---------------------------------------------------------------------------
__global__ __launch_bounds__(256)
void edge_scatter_kernel(const float* __restrict__ hs,
                         const float* __restrict__ ew,
                         const int*   __restrict__ src,
                         const int*   __restrict__ tgt,
                         float*       __restrict__ pooled,
                         int E)
{
    int tid = blockIdx.x * blockDim.x + threadIdx.x;
    int i = tid * 4;
    if (i + 4 <= E) {
        v4i s4 = __builtin_nontemporal_load(reinterpret_cast<const v4i*>(src + i));
        v4i t4 = __builtin_nontemporal_load(reinterpret_cast<const v4i*>(tgt + i));
        v4f w4 = __builtin_nontemporal_load(reinterpret_cast<const v4f*>(ew + i));
#pragma unroll
        for (int j = 0; j < 4; ++j) {
            v2f h = *reinterpret_cast<const v2f*>(hs + 2 * s4[j]);
            gadd_f32(pooled + 2 * t4[j] + 0, w4[j] * h.x);
            gadd_f32(pooled + 2 * t4[j] + 1, w4[j] * h.y);
        }
    } else {
        for (int e = i; e < E; ++e) {
            int sj = src[e], tj = tgt[e];
            float wj = ew[e];
            v2f h = *reinterpret_cast<const v2f*>(hs + 2 * sj);
            gadd_f32(pooled + 2 * tj + 0, wj * h.x);
            gadd_f32(pooled + 2 * tj + 1, wj * h.y);
        }
    }
}

// ---------------------------------------------------------------------------
// Pass 2: out = hidden + pooled @ W via V_WMMA_F32_16X16X4_F32.
// Each wave handles 32 nodes per WMMA:
//   B (4x16): lanes 0-15  -> K rows 0,1 = pooled[base+lane].{x,y}   (tile 0)
//             lanes 16-31 -> K rows 2,3 = pooled[base+lane].{x,y}   (tile 1)
//   A (16x4): rows 0,1 carry W^T acting on K=0,1; rows 2,3 carry W^T on K=2,3.
//   D rows 0,1 = tile0 result; rows 2,3 = tile1 result -> d[0..3] on lanes 0-15.
// Wave-uniform early-exit keeps EXEC all-ones at the WMMA.
// ---------------------------------------------------------------------------
__global__ __launch_bounds__(256)
void finalize_wmma_kernel(const float* __restrict__ hs,
                          const float* __restrict__ pooled,
                          const float* __restrict__ Wp,
                          float*       __restrict__ out,
                          int N)
{
    const int lane = threadIdx.x & 31;
    const int wave = threadIdx.x >> 5;
    const int tile = blockIdx.x * (blockDim.x >> 5) + wave;
    const int numTiles = (N + 31) >> 5;
    if (tile >= numTiles) return;          // uniform across the wave

    const int base = tile * 32;

    // B operand: pooled float2 for this lane's node (zero beyond N).
    int  nidx = base + lane;
    bool nv   = nidx < N;
    int  ci   = nv ? nidx : 0;
    v2f b;
    b.x = nv ? pooled[2 * ci + 0] : 0.0f;
    b.y = nv ? pooled[2 * ci + 1] : 0.0f;

    // A operand: A[m, k0+j] = W[j][r] with r = m - 2*half, nonzero for r in {0,1}.
    int  r  = (lane & 15) - 2 * (lane >> 4);
    bool av = (r >= 0) && (r < 2);
    int  rc = av ? r : 0;
    v2f a;
    a.x = av ? Wp[rc]     : 0.0f;   // W[0][r]
    a.y = av ? Wp[2 + rc] : 0.0f;   // W[1][r]

    v8f c = {};
    v8f d = __builtin_amdgcn_wmma_f32_16x16x4_f32(
        /*neg_a=*/false, a, /*neg_b=*/false, b,
        /*c_mod=*/(short)0, c, /*reuse_a=*/false, /*reuse_b=*/false);

    if (lane < 16) {
        int n0 = base + lane;
        int n1 = base + 16 + lane;
        if (n0 < N) {
            v2f h0 = *reinterpret_cast<const v2f*>(hs + 2 * n0);
            v2f o0; o0.x = h0.x + d[0]; o0.y = h0.y + d[1];
            *reinterpret_cast<v2f*>(out + 2 * n0) = o0;
        }
        if (n1 < N) {
            v2f h1 = *reinterpret_cast<const v2f*>(hs + 2 * n1);
            v2f o1; o1.x = h1.x + d[2]; o1.y = h1.y + d[3];
            *reinterpret_cast<v2f*>(out + 2 * n1) = o1;
        }
    }
}

extern "C" void kernel_launch(void* const* d_in, const int* in_sizes, int n_in,
                              void* d_out, int out_size, void* d_ws, size_t ws_size,
                              hipStream_t stream) {
    const float* hs  = (const float*)d_in[0];   // hidden_state [N,2]
    const float* ew  = (const float*)d_in[1];   // edge_weight  [E,1]
    const int*   src = (const int*)  d_in[2];   // [E]
    const int*   tgt = (const int*)  d_in[3];   // [E]
    const float* Wp  = (const float*)d_in[4];   // [2,2] row-major
    float*       out = (float*)d_out;

    const int N = in_sizes[0] / 2;
    const int E = in_sizes[2];

    const size_t pooledBytes = (size_t)N * 2 * sizeof(float);
    // Prefer scratch; fall back to accumulating in d_out (finalize's per-wave
    // disjoint 32-node windows make the in-place read-modify-write safe).
    float* pooled = (ws_size >= pooledBytes) ? (float*)d_ws : out;

    hipMemsetAsync(pooled, 0, pooledBytes, stream);

    const int edgeThreads = (E + 3) / 4;
    const int edgeBlocks  = (edgeThreads + 255) / 256;
    edge_scatter_kernel<<<edgeBlocks, 256, 0, stream>>>(hs, ew, src, tgt, pooled, E);

    const int numTiles = (N + 31) / 32;
    const int finBlocks = (numTiles + 7) / 8;   // 8 waves per 256-thread block
    finalize_wmma_kernel<<<finBlocks, 256, 0, stream>>>(hs, pooled, Wp, out, N);
}